// Decoding_17660905521232
// MI455X (gfx1250) — compile-verified
//
#include <hip/hip_runtime.h>
#include <math.h>

#define KBINS 128
#define KP1   129
#define KPAD  132            // padded row length: 132*4B = 528B, multiple of 16B
#define NLAT  16
#define N_TOTAL 100000.0

typedef float v2f __attribute__((ext_vector_type(2)));
typedef float v4f __attribute__((ext_vector_type(4)));
typedef float v8f __attribute__((ext_vector_type(8)));

// ---------------------------------------------------------------- utilities

__device__ __forceinline__ float wave_max32(float v) {
    #pragma unroll
    for (int off = 16; off >= 1; off >>= 1)
        v = fmaxf(v, __shfl_xor(v, off, 32));
    return v;
}

__device__ __forceinline__ float wave_sum32(float v) {
    #pragma unroll
    for (int off = 16; off >= 1; off >>= 1)
        v += __shfl_xor(v, off, 32);
    return v;
}

// ---------------------------------------------------------------- kernel 0: zero accumulators

__global__ void init_acc_kernel(double* acc) {
    if (threadIdx.x == 0) { acc[0] = 0.0; acc[1] = 0.0; }
}

// ---------------------------------------------------------------- kernel 1: V[r,g,k] = exp(uh[g,k] + sum_l reflatent[r,l]*lw[g,l,k])
// One wave per gene; the 16x16 x 16x129 product runs on V_WMMA_F32_16X16X4_F32.
__global__ __launch_bounds__(128) void einsum_wmma_kernel(
    const float* __restrict__ reflatent,   // [16,16]
    const float* __restrict__ lw,          // [G,16,129]
    const float* __restrict__ uh,          // [G,129]
    float* __restrict__ V,                 // [16,G,KPAD]
    int n_genes)
{
    const int lane = threadIdx.x & 31;
    const int wave = threadIdx.x >> 5;
    const int g = blockIdx.x * 4 + wave;
    if (g >= n_genes) return;              // uniform per-wave
    const int half = lane >> 4;            // 0: lanes 0-15, 1: lanes 16-31
    const int m    = lane & 15;

    // A operands: 16x4 f32 layout -> v.x=K(0|2), v.y=K(1|3) per half
    v2f a[4];
    #pragma unroll
    for (int kc = 0; kc < 4; ++kc) {
        const int kb = kc * 4 + half * 2;
        a[kc].x = reflatent[m * NLAT + kb];
        a[kc].y = reflatent[m * NLAT + kb + 1];
    }

    const float* lwg = lw + (size_t)g * (NLAT * KP1);
    const float* uhg = uh + (size_t)g * KP1;

    for (int t = 0; t < 8; ++t) {          // 8 full 16-column tiles (cols 0..127)
        const int c = t * 16 + m;
        v8f acc = {};
        #pragma unroll
        for (int kc = 0; kc < 4; ++kc) {
            const int kb = kc * 4 + half * 2;
            v2f b;
            b.x = lwg[(size_t)kb * KP1 + c];
            b.y = lwg[(size_t)(kb + 1) * KP1 + c];
            acc = __builtin_amdgcn_wmma_f32_16x16x4_f32(
                false, a[kc], false, b, (short)0, acc, false, false);
        }
        const float uhv = uhg[c];          // shared by all 8 rows in this lane
        // C/D layout: VGPR j -> row (j + 8*half), col = lane&15
        #pragma unroll
        for (int j = 0; j < 8; ++j) {
            const int r = j + half * 8;
            V[((size_t)r * n_genes + g) * KPAD + c] = __expf(uhv + acc[j]);
        }
    }

    // last column c = 128 (scalar path, lanes 0-15 each handle one r)
    if (half == 0) {
        float s = 0.f;
        #pragma unroll
        for (int l = 0; l < NLAT; ++l)
            s += reflatent[m * NLAT + l] * lwg[(size_t)l * KP1 + KBINS];
        V[((size_t)m * n_genes + g) * KPAD + KBINS] = __expf(uhg[KBINS] + s);
    }
}

// ---------------------------------------------------------------- kernel 2: per-gene width softmax + cumulative edges
// One wave per gene; lane owns bins k = 4*lane..4*lane+3. Aligned b128 loads/stores.
__global__ __launch_bounds__(128) void widths_kernel(
    const float* __restrict__ uw,          // [G,128]
    float* __restrict__ wout,              // [G,128] normalized widths
    float* __restrict__ cw,                // [G,KPAD] cumulative edges, cw[128]=1
    int n_genes)
{
    const int lane = threadIdx.x & 31;
    const int wave = threadIdx.x >> 5;
    const int g = blockIdx.x * 4 + wave;
    if (g >= n_genes) return;

    const v4f u = *(const v4f*)(uw + (size_t)g * KBINS + 4 * lane);

    float mw = fmaxf(fmaxf(u.x, u.y), fmaxf(u.z, u.w));
    mw = wave_max32(mw);

    float e[4];
    float lane_sum = 0.f;
    #pragma unroll
    for (int j = 0; j < 4; ++j) { e[j] = __expf(u[j] - mw); lane_sum += e[j]; }

    float incl = lane_sum;
    #pragma unroll
    for (int off = 1; off < 32; off <<= 1) {
        float nb = __shfl_up(incl, off, 32);
        if (lane >= off) incl += nb;
    }
    const float excl = incl - lane_sum;
    const float S    = __shfl(incl, 31, 32);
    const float invS = 1.0f / S;

    v4f wv, cv;
    float p = excl;
    #pragma unroll
    for (int j = 0; j < 4; ++j) {
        cv[j] = p * invS;
        wv[j] = e[j] * invS;
        p += e[j];
    }
    *(v4f*)(wout + (size_t)g * KBINS + 4 * lane) = wv;
    *(v4f*)(cw   + (size_t)g * KPAD  + 4 * lane) = cv;
    if (lane == 31) cw[(size_t)g * KPAD + KBINS] = 1.0f;
}

// ---------------------------------------------------------------- kernel 3: lognorm[r,g] = log(sum_k 0.5*(v_k+v_{k+1})*w_k)
// One wave per (r,g) row.
__global__ __launch_bounds__(256) void lognorm_kernel(
    const float* __restrict__ V,           // [16,G,KPAD]
    const float* __restrict__ w,           // [G,128]
    float* __restrict__ lognorm,           // [16*G]
    int n_genes)
{
    const int lane = threadIdx.x & 31;
    const int wave = threadIdx.x >> 5;
    const int id = blockIdx.x * 8 + wave;  // id = r*G + g
    if (id >= NLAT * n_genes) return;
    const int g = id % n_genes;

    const float* Vrow = V + (size_t)id * KPAD;
    const v4f v  = *(const v4f*)(Vrow + 4 * lane);
    const v4f wv = *(const v4f*)(w + (size_t)g * KBINS + 4 * lane);

    float vnext = __shfl_down(v.x, 1, 32);
    if (lane == 31) vnext = Vrow[KBINS];

    float ns = 0.f;
    #pragma unroll
    for (int j = 0; j < 4; ++j) {
        const float vk1 = (j < 3) ? v[j + 1] : vnext;
        ns += 0.5f * (v[j] + vk1) * wv[j];
    }
    const float norm = wave_sum32(ns);
    if (lane == 0) lognorm[id] = __logf(norm);
}

// ---------------------------------------------------------------- kernel 4: per-cut logprob (one thread per cut)
// Binary search the cached cw row; everything else is precomputed.
__global__ __launch_bounds__(256) void cuts_kernel(
    const float* __restrict__ coords,
    const int*   __restrict__ ridx,
    const int*   __restrict__ gidx,
    const float* __restrict__ V,           // [16,G,KPAD]
    const float* __restrict__ w,           // [G,128]
    const float* __restrict__ cw,          // [G,KPAD]
    const float* __restrict__ lognorm,     // [16*G]
    int n_cuts, int n_genes,
    double* __restrict__ acc)
{
    __shared__ double blk;
    if (threadIdx.x == 0) blk = 0.0;
    __syncthreads();

    const int cut = blockIdx.x * blockDim.x + threadIdx.x;
    float lp = 0.f;
    if (cut < n_cuts) {
        const int g = gidx[cut];
        const int r = ridx[cut];
        float x = coords[cut];
        x = fminf(fmaxf(x, 0.0f), 1.0f - 1e-6f);

        const float* cwrow = cw + (size_t)g * KPAD;
        // invariant: cw[lo] <= x < cw[hi]; cw[0]=0, cw[128]=1 > x
        int lo = 0, hi = KBINS;
        #pragma unroll
        for (int i = 0; i < 7; ++i) {
            const int mid = (lo + hi) >> 1;
            if (cwrow[mid] <= x) lo = mid; else hi = mid;
        }
        const int k = lo;                       // in [0,127]
        const float wk    = w[(size_t)g * KBINS + k];
        const float alpha = (x - cwrow[k]) / wk;
        const float* Vrow = V + ((size_t)r * n_genes + g) * KPAD;
        const float vk  = Vrow[k];
        const float vk1 = Vrow[k + 1];
        lp = __logf(vk + alpha * (vk1 - vk)) - lognorm[(size_t)r * n_genes + g];
    }

    lp = wave_sum32(lp);
    if ((threadIdx.x & 31) == 0) atomicAdd(&blk, (double)lp);
    __syncthreads();
    if (threadIdx.x == 0) atomicAdd(acc, blk);
}

// ---------------------------------------------------------------- fallback: fully inline wave-per-cut spline (no workspace)

__device__ __forceinline__ float delta_inline(const float* reflatent, const float* lwg,
                                              int r, int k)
{
    float s = 0.f;
    const float* rr = reflatent + r * NLAT;
    #pragma unroll
    for (int l = 0; l < NLAT; ++l)
        s += rr[l] * lwg[(size_t)l * KP1 + k];
    return s;
}

__global__ __launch_bounds__(256) void spline_fallback_kernel(
    const float* __restrict__ coords,
    const int*   __restrict__ ridx,
    const int*   __restrict__ gidx,
    const float* __restrict__ heights,     // [G,129]
    const float* __restrict__ widths,      // [G,128]
    const float* __restrict__ reflatent,   // [16,16]
    const float* __restrict__ lw,          // [G,16,129]
    int n_cuts, int n_genes,
    double* __restrict__ acc)
{
    __shared__ double blk;
    if (threadIdx.x == 0) blk = 0.0;
    __syncthreads();

    const int lane = threadIdx.x & 31;
    const int wave = threadIdx.x >> 5;
    const int cut  = blockIdx.x * 8 + wave;

    if (cut < n_cuts) {
        const int g = gidx[cut];
        const int r = ridx[cut];
        float x = coords[cut];
        x = fminf(fmaxf(x, 0.0f), 1.0f - 1e-6f);

        const float* uw_row = widths  + (size_t)g * KBINS;
        const float* uh_row = heights + (size_t)g * KP1;
        const float* lwg    = lw + (size_t)g * (NLAT * KP1);

        const v4f u = *(const v4f*)(uw_row + 4 * lane);
        float mw = fmaxf(fmaxf(u.x, u.y), fmaxf(u.z, u.w));
        mw = wave_max32(mw);

        float e[4];
        float lane_sum = 0.f;
        #pragma unroll
        for (int j = 0; j < 4; ++j) { e[j] = __expf(u[j] - mw); lane_sum += e[j]; }

        float incl = lane_sum;
        #pragma unroll
        for (int off = 1; off < 32; off <<= 1) {
            float nb = __shfl_up(incl, off, 32);
            if (lane >= off) incl += nb;
        }
        const float excl = incl - lane_sum;
        const float S    = __shfl(incl, 31, 32);
        const float invS = 1.0f / S;

        float p[4];
        p[0] = excl; p[1] = p[0] + e[0]; p[2] = p[1] + e[1]; p[3] = p[2] + e[2];

        float vv[4];
        #pragma unroll
        for (int j = 0; j < 4; ++j) {
            const int k = 4 * lane + j;
            vv[j] = __expf(uh_row[k] + delta_inline(reflatent, lwg, r, k));
        }
        float vnext = __shfl_down(vv[0], 1, 32);
        if (lane == 31)
            vnext = __expf(uh_row[KBINS] + delta_inline(reflatent, lwg, r, KBINS));

        float ns = 0.f;
        #pragma unroll
        for (int j = 0; j < 4; ++j) {
            const float vk1 = (j < 3) ? vv[j + 1] : vnext;
            ns += 0.5f * (vv[j] + vk1) * (e[j] * invS);
        }
        const float norm = wave_sum32(ns);

        float lp = 0.f;
        #pragma unroll
        for (int j = 0; j < 4; ++j) {
            const float c0 = p[j] * invS;
            const float c1 = (p[j] + e[j]) * invS;
            const bool last = (lane == 31) && (j == 3);
            if (x >= c0 && (x < c1 || last)) {
                const float alpha = (x - c0) / (e[j] * invS);
                const float vk1 = (j < 3) ? vv[j + 1] : vnext;
                lp = __logf(vv[j] + alpha * (vk1 - vv[j])) - __logf(norm);
            }
        }
        lp = wave_sum32(lp);
        if (lane == 0) atomicAdd(&blk, (double)lp);
    }

    __syncthreads();
    if (threadIdx.x == 0) atomicAdd(acc, blk);
}

// ---------------------------------------------------------------- kernel 5: sum of logit_weight^2
__global__ __launch_bounds__(256) void kl_sumsq_kernel(
    const float* __restrict__ lw, long long n, double* __restrict__ acc)  // acc[1]
{
    __shared__ double blk;
    if (threadIdx.x == 0) blk = 0.0;
    __syncthreads();

    double s = 0.0;
    const long long stride = (long long)gridDim.x * blockDim.x;
    for (long long i = (long long)blockIdx.x * blockDim.x + threadIdx.x; i < n; i += stride) {
        const float v = lw[i];
        s += (double)v * (double)v;
    }
    #pragma unroll
    for (int off = 16; off >= 1; off >>= 1)
        s += __shfl_xor(s, off, 32);
    if ((threadIdx.x & 31) == 0) atomicAdd(&blk, s);
    __syncthreads();
    if (threadIdx.x == 0) atomicAdd(acc + 1, blk);
}

// ---------------------------------------------------------------- kernel 6: finalize
__global__ void finalize_kernel(const double* __restrict__ acc,
                                const int* __restrict__ n_cells_p,
                                long long kl_count,
                                float* __restrict__ out)
{
    if (threadIdx.x == 0 && blockIdx.x == 0) {
        const double likelihood = acc[0] * (N_TOTAL / (double)n_cells_p[0]);
        const double C0 = -0.9189385332046727;          // -0.5*log(2*pi), sigma = 1
        const double kl = (double)kl_count * C0 - 0.5 * acc[1];
        out[0] = (float)((-likelihood - kl) / N_TOTAL);
    }
}

// ---------------------------------------------------------------- launch

extern "C" void kernel_launch(void* const* d_in, const int* in_sizes, int n_in,
                              void* d_out, int out_size, void* d_ws, size_t ws_size,
                              hipStream_t stream) {
    const float* coords    = (const float*)d_in[0];
    const int*   ridx      = (const int*)  d_in[1];
    const int*   gidx      = (const int*)  d_in[2];
    const int*   ncells    = (const int*)  d_in[3];
    const float* reflatent = (const float*)d_in[4];
    const float* lw        = (const float*)d_in[5];
    const float* uw        = (const float*)d_in[6];
    const float* uh        = (const float*)d_in[7];

    const int n_cuts  = in_sizes[0];
    const int n_genes = in_sizes[6] / KBINS;
    const long long kl_count = (long long)n_genes * NLAT * KP1;

    double* acc = (double*)d_ws;

    // workspace layout (fast path)
    const size_t V_bytes  = (size_t)NLAT * n_genes * KPAD * sizeof(float);
    const size_t cw_bytes = (size_t)n_genes * KPAD * sizeof(float);
    const size_t w_bytes  = (size_t)n_genes * KBINS * sizeof(float);
    const size_t ln_bytes = (size_t)NLAT * n_genes * sizeof(float);
    const size_t need = 64 + V_bytes + cw_bytes + w_bytes + ln_bytes;

    init_acc_kernel<<<1, 64, 0, stream>>>(acc);

    if (ws_size >= need) {
        char* base = (char*)d_ws + 64;
        float* V       = (float*)base;
        float* cw      = (float*)(base + V_bytes);
        float* wbuf    = (float*)(base + V_bytes + cw_bytes);
        float* lognorm = (float*)(base + V_bytes + cw_bytes + w_bytes);

        const int gblocks = (n_genes + 3) / 4;
        einsum_wmma_kernel<<<gblocks, 128, 0, stream>>>(reflatent, lw, uh, V, n_genes);
        widths_kernel<<<gblocks, 128, 0, stream>>>(uw, wbuf, cw, n_genes);

        const int nrows = NLAT * n_genes;
        lognorm_kernel<<<(nrows + 7) / 8, 256, 0, stream>>>(V, wbuf, lognorm, n_genes);

        cuts_kernel<<<(n_cuts + 255) / 256, 256, 0, stream>>>(
            coords, ridx, gidx, V, wbuf, cw, lognorm, n_cuts, n_genes, acc);
    } else {
        spline_fallback_kernel<<<(n_cuts + 7) / 8, 256, 0, stream>>>(
            coords, ridx, gidx, uh, uw, reflatent, lw, n_cuts, n_genes, acc);
    }

    kl_sumsq_kernel<<<2048, 256, 0, stream>>>(lw, kl_count, acc);

    finalize_kernel<<<1, 64, 0, stream>>>(acc, ncells, kl_count, (float*)d_out);
}